// InversePenaltyTracker_43808666419798
// MI455X (gfx1250) — compile-verified
//
#include <hip/hip_runtime.h>

// MI455X / gfx1250, wave32. Blocked Sherman-Morrison via f32 WMMA (16x16x4).

typedef float v2f __attribute__((ext_vector_type(2)));
typedef float v8f __attribute__((ext_vector_type(8)));

#define NBATCH   1024
#define DDIM     128
#define RRANK    64
#define RBLK     16

#define LDA 132   // padded stride for A   (even -> b64-aligned)
#define LDU 132   // padded stride for U   (even)
#define SW  17    // stride for W (scalar access only)
#define SZ  18    // stride for Z / Zscaled (even -> b64-aligned)
#define SM  17    // stride for M,G,C (scalar access only)

// LDS layout (float offsets)
#define OFF_A   0
#define OFF_U   (OFF_A  + 128*LDA)   // 16896
#define OFF_W   (OFF_U  + 16*LDU)    // +2112
#define OFF_Z   (OFF_W  + 128*SW)    // +2176
#define OFF_ZS  (OFF_Z  + 128*SZ)    // +2304
#define OFF_M   (OFF_ZS + 128*SZ)    // +2304
#define OFF_G   (OFF_M  + 16*SM)     // +272
#define OFF_C   (OFF_G  + 16*SM)     // +272
#define OFF_EPS (OFF_C  + 16*SM)     // +272
#define OFF_NID (OFF_EPS + 16)
#define OFF_ET  (OFF_NID + 16)
#define LDS_FLOATS (OFF_ET + 4)      // 26644 floats = 106576 bytes

#define STAB_EPS_F     1.0e-6f
#define PERIODIC_EPS_F 1.0e-5f

static __device__ __forceinline__ v8f wmma_f32_4(v2f a, v2f b, v8f c) {
  return __builtin_amdgcn_wmma_f32_16x16x4_f32(false, a, false, b, (short)0, c,
                                               false, false);
}

__global__ __launch_bounds__(128)
void InversePenaltyTracker_43808666419798_kernel(const float* __restrict__ A0,
                                                 const float* __restrict__ Uin,
                                                 float* __restrict__ Out) {
  extern __shared__ float lds[];
  float* As   = lds + OFF_A;    // A (128 x 128), stride LDA
  float* Us   = lds + OFF_U;    // U block (16 x 128), stride LDU
  float* Ws   = lds + OFF_W;    // W = A*U^T (128 x 16), stride SW
  float* Zm   = lds + OFF_Z;    // Z (128 x 16), stride SZ
  float* Zc   = lds + OFF_ZS;   // Z * diag(-1/delta), stride SZ
  float* Ms   = lds + OFF_M;    // M = U A U^T (16x16), stride SM
  float* Gs   = lds + OFF_G;    // G = U U^T   (16x16), stride SM
  float* Cs   = lds + OFF_C;    // coefficients c_ij (16x16), stride SM
  float* epsA = lds + OFF_EPS;  // eps before step i (16)
  float* nid  = lds + OFF_NID;  // -1/delta_i (0 if skipped) (16)
  float* epst = lds + OFF_ET;   // total eps added this block

  const int tid   = threadIdx.x;   // 0..127
  const int lane  = tid & 31;
  const int wave  = tid >> 5;      // 0..3
  const int n16   = lane & 15;
  const int kh    = lane >> 4;     // half-wave select (K pair)
  const int batch = blockIdx.x;

  const float* Ab = A0  + (size_t)batch * DDIM * DDIM;
  const float* Ub = Uin + (size_t)batch * RRANK * DDIM;
  float*       Ob = Out + (size_t)batch * DDIM * DDIM;

  // ---- load A0 into LDS (coalesced, row per iteration) ----
  for (int r = 0; r < 128; ++r) As[r * LDA + tid] = Ab[r * 128 + tid];
  __syncthreads();

  for (int b = 0; b < 4; ++b) {
    // ---- load U panel for this block of 16 updates ----
    for (int i = 0; i < 16; ++i)
      Us[i * LDU + tid] = Ub[(16 * b + i) * 128 + tid];
    if (b < 3)  // hint next panel toward L2 (global_prefetch_b8)
      __builtin_prefetch(&Ub[(16 * (b + 1)) * 128 + tid], 0, 1);
    __syncthreads();

    // ---- GEMM1: W(128x16) = A(128x128) x U^T(128x16), K=128 [WMMA] ----
    for (int mt = wave; mt < 8; mt += 4) {
      v8f acc = {0.f, 0.f, 0.f, 0.f, 0.f, 0.f, 0.f, 0.f};
      const int m = mt * 16 + n16;
      #pragma unroll 8
      for (int kk = 0; kk < 32; ++kk) {
        const int k0 = kk * 4;
        v2f a  = *(const v2f*)&As[m * LDA + k0 + 2 * kh];       // A[m][k..k+1]
        v2f bb = *(const v2f*)&Us[n16 * LDU + k0 + 2 * kh];     // B[k][n]=U^T
        acc = wmma_f32_4(a, bb, acc);
      }
      #pragma unroll
      for (int v = 0; v < 8; ++v)
        Ws[(mt * 16 + v + 8 * kh) * SW + n16] = acc[v];
    }
    __syncthreads();

    // ---- GEMM2: M = U x W (wave0), G = U x U^T (wave1); 16x16, K=128 ----
    if (wave < 2) {
      v8f acc = {0.f, 0.f, 0.f, 0.f, 0.f, 0.f, 0.f, 0.f};
      #pragma unroll 8
      for (int kk = 0; kk < 32; ++kk) {
        const int k0 = kk * 4;
        v2f a = *(const v2f*)&Us[n16 * LDU + k0 + 2 * kh];  // A-op: U rows
        v2f bb;
        if (wave == 0) {
          bb.x = Ws[(k0 + 2 * kh) * SW + n16];              // B[k][n] = W[k][n]
          bb.y = Ws[(k0 + 2 * kh + 1) * SW + n16];
        } else {
          bb = *(const v2f*)&Us[n16 * LDU + k0 + 2 * kh];   // B = U^T
        }
        acc = wmma_f32_4(a, bb, acc);
      }
      float* Dst = (wave == 0) ? Ms : Gs;
      #pragma unroll
      for (int v = 0; v < 8; ++v) Dst[(v + 8 * kh) * SM + n16] = acc[v];
    }
    __syncthreads();

    // ---- scalar phase (wave 0): 16x16 recurrences, exact per-step
    //      delta / stability / periodic-eps semantics ----
    if (wave == 0) {
      const int i = n16;
      float Mrow[16], Grow[16];
      #pragma unroll
      for (int k = 0; k < 16; ++k) {
        Mrow[k] = Ms[i * SM + k];
        Grow[k] = Gs[i * SM + k];
      }
      float t[16], invd[16];
      float eps_cur = 0.f, eps_tot = 0.f;
      #pragma unroll
      for (int j = 0; j < 16; ++j) {
        if (lane == 0) epsA[j] = eps_cur;
        // t_ij = u_i . z_j = (M_ij + eps_j*G_ij) - sum_{k<j} c_jk * t_ik
        float acc = Mrow[j] + eps_cur * Grow[j];
        #pragma unroll
        for (int k = 0; k < j; ++k) {
          float tjk = __shfl(t[k], j, 32);        // row j of t, from lane j
          acc -= (tjk * invd[k]) * t[k];
        }
        t[j] = acc;
        float tjj = __shfl(t[j], j, 32);
        float dj  = 1.0f + tjj;
        float adj = fabsf(dj);
        bool unstable = !(adj >= STAB_EPS_F && adj <= 3.0e38f); // |d|<eps, NaN, Inf
        float iv = unstable ? 0.0f : (1.0f / dj);
        invd[j] = iv;
        if (lane < 16) Cs[i * SM + j] = t[j] * iv;   // c_ij (skipped -> 0)
        if (lane == 0) nid[j] = -iv;                 // GEMM4 column scale
        if (unstable) { eps_cur += STAB_EPS_F; eps_tot += STAB_EPS_F; }
        if (((16 * b + j + 1) % 50) == 0) {          // periodic stabilization
          eps_cur += PERIODIC_EPS_F; eps_tot += PERIODIC_EPS_F;
        }
      }
      if (lane == 0) epst[0] = eps_tot;
    }
    __syncthreads();

    // ---- Z phase: per-row 16-deep recurrence, z = w' - C z (lower-tri) ----
    {
      const int d = tid;
      float w[16], z[16];
      #pragma unroll
      for (int ii = 0; ii < 16; ++ii)
        w[ii] = Ws[d * SW + ii] + epsA[ii] * Us[ii * LDU + d];
      #pragma unroll
      for (int ii = 0; ii < 16; ++ii) {
        float acc = w[ii];
        #pragma unroll
        for (int j = 0; j < ii; ++j) acc -= Cs[ii * SM + j] * z[j];
        z[ii] = acc;
      }
      #pragma unroll
      for (int ii = 0; ii < 16; ++ii) {
        Zm[d * SZ + ii] = z[ii];
        Zc[d * SZ + ii] = z[ii] * nid[ii];  // column scaled by -1/delta (0 if skipped)
      }
    }
    __syncthreads();

    // ---- GEMM4: A(128x128) += Zc(128x16) x Z^T(16x128), K=16 [WMMA] ----
    for (int tix = wave; tix < 64; tix += 4) {
      const int mt = tix >> 3, nt = tix & 7;
      v8f acc;
      #pragma unroll
      for (int v = 0; v < 8; ++v)
        acc[v] = As[(mt * 16 + v + 8 * kh) * LDA + nt * 16 + n16];
      #pragma unroll
      for (int kk = 0; kk < 4; ++kk) {
        const int k0 = kk * 4;
        v2f a  = *(const v2f*)&Zc[(mt * 16 + n16) * SZ + k0 + 2 * kh];
        v2f bb = *(const v2f*)&Zm[(nt * 16 + n16) * SZ + k0 + 2 * kh]; // B[k][n]=Z[n][k]
        acc = wmma_f32_4(a, bb, acc);
      }
      #pragma unroll
      for (int v = 0; v < 8; ++v)
        As[(mt * 16 + v + 8 * kh) * LDA + nt * 16 + n16] = acc[v];
    }
    __syncthreads();

    // ---- fold accumulated eps into the diagonal ----
    As[tid * LDA + tid] += epst[0];
    __syncthreads();
  }

  // ---- write back ----
  for (int r = 0; r < 128; ++r) Ob[r * 128 + tid] = As[r * LDA + tid];
}

extern "C" void kernel_launch(void* const* d_in, const int* in_sizes, int n_in,
                              void* d_out, int out_size, void* d_ws, size_t ws_size,
                              hipStream_t stream) {
  const float* A0 = (const float*)d_in[0];   // (B,128,128) f32
  const float* u  = (const float*)d_in[1];   // (B,64,128) f32
  float* out = (float*)d_out;                // (B,128,128) f32
  (void)in_sizes; (void)n_in; (void)out_size; (void)d_ws; (void)ws_size;

  dim3 grid(NBATCH), block(128);
  size_t shmem = (size_t)LDS_FLOATS * sizeof(float);  // ~104 KB dynamic LDS
  hipLaunchKernelGGL(InversePenaltyTracker_43808666419798_kernel,
                     grid, block, shmem, stream, A0, u, out);
}